// DotMatrix_69561290326706
// MI455X (gfx1250) — compile-verified
//
#include <hip/hip_runtime.h>

// CDNA5 / gfx1250 (MI455X): wave32, WMMA 16x16 tiles.
//
// Reference: for each ell in {0,1,2,3} (m = 2*ell+1) and each (b, t):
//   p2[j,m,:]  = part[j, msize-1-m, :] * (-1)^(m-ell)
//   out_r[i,j] = sum_m ( a_r[i,m]*p2_r[j,m] - a_i[i,m]*p2_i[j,m] )
//   out_i[i,j] = sum_m ( a_r[i,m]*p2_i[j,m] + a_i[i,m]*p2_r[j,m] )
// Output [2,256,256,128,2] f32 = 134 MB  -> kernel is output-BW bound
// (23.3 TB/s => ~5.8us floor). Inputs are 2 MB total (L2 resident).
// Compute is 0.54 GFLOP; keep it in native f32 via V_WMMA_F32_16X16X4_F32.
//
// WMMA f32 16x16x4 operand layout (ISA 7.12.2):
//   A (16x4): lane L -> row M = L&15; VGPR0 holds K = 4s + 2*(L>>4),
//             VGPR1 holds K+1.  With K packed as (m,c) pairs (c = re/im),
//             each lane's (K,K+1) pair == one contiguous float2 in memory.
//   B (4x16): symmetric transpose: lane L -> col N = L&15; VGPR0 holds
//             row K = 4s + 2*(L>>4), VGPR1 row K+1.
//   C/D (16x16 f32): VGPR v, lane L -> M = v + 8*(L>>4), N = L&15.
//
// ell / msize / S (K-step count) are wave-uniform; they are forced into
// SGPRs via readfirstlane so the K loop is a *uniform* s_cbranch loop and
// EXEC is provably all-1s at every WMMA (ISA 7.12 requirement).

typedef __attribute__((ext_vector_type(2))) float v2f;
typedef __attribute__((ext_vector_type(8))) float v8f;

struct Reps { const float* r[4]; };

__global__ __launch_bounds__(256) void DotMatrix_wmma_kernel(Reps reps,
                                                             float* __restrict__ out) {
  constexpr int N   = 256;
  constexpr int TAU = 32;
  constexpr int TCH = 128;   // 4 * TAU concatenated channels

  const int blk = blockIdx.x;        // 512 blocks: b * 256 + ti * 16 + tj
  const int b   = blk >> 8;
  const int ti  = (blk >> 4) & 15;
  const int tj  = blk & 15;

  const int tid  = threadIdx.x;
  const int lane = tid & 31;
  const int h    = lane >> 4;        // lane half (selects K pair / M high rows)
  const int col  = lane & 15;

  // Wave id is uniform across the wave: pin it (and everything derived from
  // it) to SGPRs so control flow stays uniform.
  const int wave  = __builtin_amdgcn_readfirstlane(tid >> 5);  // 0..7
  const int ell   = wave >> 1;       // wave w owns t in [16w,16w+16) -> one ell
  const int msize = 2 * ell + 1;     // 1,3,5,7
  const int S     = (msize + 1) >> 1;// ceil(2*msize / 4) WMMA K-steps: 1..4
  const float* __restrict__ rep = reps.r[ell];

  const int i_row = ti * 16 + col;   // A row for this lane
  const int j_row = tj * 16 + col;   // B col for this lane

  for (int iter = 0; iter < 16; ++iter) {
    const int t  = wave * 16 + iter; // global output channel 0..127 (uniform)
    const int tl = t & 31;           // channel within this ell's tau block

    // rep element [b, row, tl, m, c] at (((b*N+row)*TAU+tl)*msize + m)*2 + c
    const float* repA = rep + (long)(((b * N + i_row) * TAU + tl) * msize) * 2;
    const float* repB = rep + (long)(((b * N + j_row) * TAU + tl) * msize) * 2;

    v8f cr = {};                     // real accumulator (16x16 f32 tile)
    v8f ci = {};                     // imag accumulator

    for (int s = 0; s < S; ++s) {    // S in SGPR -> uniform loop, EXEC full
      const int   m     = 2 * s + h;               // this lane's m index
      const int   m_eff = (m < msize) ? m : (msize - 1);  // clamp (branch-free)
      const float valid = (m < msize) ? 1.0f : 0.0f;      // v_cndmask, no EXEC div
      const float sgn   = ((m_eff + ell) & 1) ? -1.0f : 1.0f; // (-1)^(m-ell)

      // A pair (re, im) for row i, channel tl, index m
      v2f a = *(const v2f*)(repA + m_eff * 2);
      // p2 pair: flipped m index, signed
      v2f p = *(const v2f*)(repB + (msize - 1 - m_eff) * 2);
      a *= valid;
      p *= (valid * sgn);

      const v2f br = { p.x, -p.y };  // real-part B rows: (p2_r, -p2_i)
      const v2f bi = { p.y,  p.x };  // imag-part B rows: (p2_i,  p2_r)

      // (neg_a, A, neg_b, B, c_mod, C, reuse_a, reuse_b)
      cr = __builtin_amdgcn_wmma_f32_16x16x4_f32(false, a, false, br,
                                                 (short)0, cr, false, false);
      ci = __builtin_amdgcn_wmma_f32_16x16x4_f32(false, a, false, bi,
                                                 (short)0, ci, false, false);
    }

    // Store (re, im) as float2; out[b, i, j, t, c].
    // Each 64B line (8 consecutive t) is filled by this wave over 8
    // consecutive iters -> merges in the 192MB write-back L2, full-line
    // HBM evictions.
#pragma unroll
    for (int v = 0; v < 8; ++v) {
      const int  i   = ti * 16 + v + 8 * h;             // C-layout row
      const long off = (((long)(b * N + i) * N + j_row) * TCH + t) * 2;
      v2f val = { cr[v], ci[v] };
      *(v2f*)(out + off) = val;
    }
  }
}

extern "C" void kernel_launch(void* const* d_in, const int* in_sizes, int n_in,
                              void* d_out, int out_size, void* d_ws, size_t ws_size,
                              hipStream_t stream) {
  (void)in_sizes; (void)n_in; (void)out_size; (void)d_ws; (void)ws_size;
  Reps reps;
  for (int i = 0; i < 4; ++i) reps.r[i] = (const float*)d_in[i];
  float* out = (float*)d_out;
  // 512 blocks = 2 (b) * 16 * 16 tiles of the 256x256 pair matrix;
  // 256 threads = 8 waves, each wave owns 16 of the 128 tau channels.
  DotMatrix_wmma_kernel<<<dim3(512), dim3(256), 0, stream>>>(reps, out);
}